// Norm_10033043604048
// MI455X (gfx1250) — compile-verified
//
#include <hip/hip_runtime.h>

// GraphNorm (per-graph mean/var normalization), sorted batch_index.
// One workgroup per graph (256 WGs x 512 threads = 16 wave32 waves).
//   pass 1: sum(x), sum(x^2) per feature via V_WMMA_F32_16X16X4_F32 (ones-A trick)
//   pass 2: out = a[d]*x + b[d], coalesced float4 stream with NT stores.
// var = E[x^2] - 2*s*m^2 + s^2*m^2  (since sub = x - m*s, m = E[x]).

typedef __attribute__((ext_vector_type(2))) float v2f;
typedef __attribute__((ext_vector_type(4))) float v4f;
typedef __attribute__((ext_vector_type(8))) float v8f;

#define GN_D   128
#define GN_B   256
#define GN_EPS 1e-6f

__device__ __forceinline__ int lower_bound_i32(const int* __restrict__ a, int n, int v) {
  int lo = 0, hi = n;
  while (lo < hi) {
    int mid = (lo + hi) >> 1;
    if (a[mid] < v) lo = mid + 1; else hi = mid;
  }
  return lo;
}

__global__ __launch_bounds__(512) void graphnorm_kernel(
    const float* __restrict__ x,
    const int*   __restrict__ bidx,
    const float* __restrict__ weight,
    const float* __restrict__ bias,
    const float* __restrict__ mscale,
    float* __restrict__ out,
    int N)
{
  __shared__ int   s_se[2];
  __shared__ float s_red[2 * GN_D];   // [0..127]=sum(x), [128..255]=sum(x^2)
  __shared__ float s_ab[2 * GN_D];    // [0..127]=a, [128..255]=b

  const int g   = blockIdx.x;
  const int tid = threadIdx.x;

  if (tid < 2)        s_se[tid]  = lower_bound_i32(bidx, N, g + tid);
  if (tid < 2 * GN_D) s_red[tid] = 0.0f;
  __syncthreads();

  // Scalarize everything wave-uniform: graph bounds (same for all lanes) and
  // the wave index (uniform within a wave). Loop control becomes pure SALU,
  // EXEC is never touched -> WMMA provably runs with EXEC == all-ones.
  const int start = __builtin_amdgcn_readfirstlane(s_se[0]);
  const int end   = __builtin_amdgcn_readfirstlane(s_se[1]);
  const int wave  = __builtin_amdgcn_readfirstlane(tid >> 5);   // 0..15, SGPR

  const int lane = tid & 31;
  const int hl   = lane >> 4;       // half-wave: 0/1
  const int col  = lane & 15;       // feature within 16-wide tile
  const int tile = wave & 7;        // which 16-feature tile   (SGPR)
  const int nh   = wave >> 3;       // node-half: 0/1          (SGPR)
  const int f    = tile * 16 + col; // global feature id for pass 1

  // A = ones(16x4): result D[m,n] = sum_k B[k,n], independent of B's K-slot layout.
  v2f ones; ones[0] = 1.0f; ones[1] = 1.0f;
  v8f acc_s = {};   // column sums of x
  v8f acc_q = {};   // column sums of x^2

  // ---- Pass 1 main loop: full 8-node blocks, unconditional loads, imm offsets,
  // SGPR loop counter.
  int base = start + 8 * nh;
  const float* p = x + (size_t)(base + hl) * GN_D + f;
  for (; base + 8 <= end; base += 16) {
    const float d0 = p[0 * GN_D];
    const float d1 = p[2 * GN_D];
    const float d2 = p[4 * GN_D];
    const float d3 = p[6 * GN_D];
    __builtin_prefetch(p + 32 * GN_D, 0, 3);   // two blocks ahead
    p += 16 * GN_D;
    v2f b0; b0[0] = d0;      b0[1] = d1;
    v2f b1; b1[0] = d2;      b1[1] = d3;
    v2f q0; q0[0] = d0 * d0; q0[1] = d1 * d1;
    v2f q1; q1[0] = d2 * d2; q1[1] = d3 * d3;
    acc_s = __builtin_amdgcn_wmma_f32_16x16x4_f32(false, ones, false, b0, (short)0, acc_s, false, false);
    acc_q = __builtin_amdgcn_wmma_f32_16x16x4_f32(false, ones, false, q0, (short)0, acc_q, false, false);
    acc_s = __builtin_amdgcn_wmma_f32_16x16x4_f32(false, ones, false, b1, (short)0, acc_s, false, false);
    acc_q = __builtin_amdgcn_wmma_f32_16x16x4_f32(false, ones, false, q1, (short)0, acc_q, false, false);
  }

  // ---- Pass 1 tail: at most one partial block for this wave. Condition is
  // SGPR-uniform (scalar branch); loads use clamped addresses + cndmask zeroing.
  if (base < end) {
    const int e1 = end - 1;            // valid since base < end
    const int r0 = base + hl;
    const int r1 = r0 + 2, r2 = r0 + 4, r3 = r0 + 6;
    const int c0 = (r0 < e1) ? r0 : e1;
    const int c1 = (r1 < e1) ? r1 : e1;
    const int c2 = (r2 < e1) ? r2 : e1;
    const int c3 = (r3 < e1) ? r3 : e1;
    float d0 = x[(size_t)c0 * GN_D + f];
    float d1 = x[(size_t)c1 * GN_D + f];
    float d2 = x[(size_t)c2 * GN_D + f];
    float d3 = x[(size_t)c3 * GN_D + f];
    d0 = (r0 < end) ? d0 : 0.0f;
    d1 = (r1 < end) ? d1 : 0.0f;
    d2 = (r2 < end) ? d2 : 0.0f;
    d3 = (r3 < end) ? d3 : 0.0f;
    v2f b0; b0[0] = d0;      b0[1] = d1;
    v2f b1; b1[0] = d2;      b1[1] = d3;
    v2f q0; q0[0] = d0 * d0; q0[1] = d1 * d1;
    v2f q1; q1[0] = d2 * d2; q1[1] = d3 * d3;
    acc_s = __builtin_amdgcn_wmma_f32_16x16x4_f32(false, ones, false, b0, (short)0, acc_s, false, false);
    acc_q = __builtin_amdgcn_wmma_f32_16x16x4_f32(false, ones, false, q0, (short)0, acc_q, false, false);
    acc_s = __builtin_amdgcn_wmma_f32_16x16x4_f32(false, ones, false, b1, (short)0, acc_s, false, false);
    acc_q = __builtin_amdgcn_wmma_f32_16x16x4_f32(false, ones, false, q1, (short)0, acc_q, false, false);
  }

  // D/C layout: VGPR0 lanes 0-15 hold row M=0 (N=lane) -> the column sums.
  if (lane < 16) {
    atomicAdd(&s_red[f],        acc_s[0]);   // ds_add_f32, combines the two node-halves
    atomicAdd(&s_red[GN_D + f], acc_q[0]);
  }
  __syncthreads();

  if (tid < GN_D) {
    const int   cnt = end - start;
    const float c   = (float)(cnt > 0 ? cnt : 1);
    const float m   = s_red[tid] / c;
    const float e2  = s_red[GN_D + tid] / c;
    const float s   = mscale[tid];
    const float sm  = s * m;
    const float var = e2 - 2.0f * sm * m + sm * sm;
    const float a   = weight[tid] / sqrtf(var + GN_EPS);
    s_ab[tid]        = a;
    s_ab[GN_D + tid] = bias[tid] - sm * a;
  }
  __syncthreads();

  // ---- Pass 2: out = a*x + b. Each wave streams one full 512B row per
  // iteration (32 lanes x float4), 16 waves -> 16 rows per step. SGPR row
  // counter, NT stores so the streaming output doesn't thrash L2.
  const v4f A4 = *(const v4f*)(s_ab + 4 * lane);
  const v4f B4 = *(const v4f*)(s_ab + GN_D + 4 * lane);
  const float* px = x   + (size_t)(start + wave) * GN_D + 4 * lane;
  float*       po = out + (size_t)(start + wave) * GN_D + 4 * lane;
  for (int r = start + wave; r < end; r += 16) {
    const v4f xv = *(const v4f*)px;
    const v4f o  = xv * A4 + B4;                    // 4x v_fma_f32
    __builtin_nontemporal_store(o, (v4f*)po);       // streaming store
    px += 16 * GN_D;
    po += 16 * GN_D;
  }
}

extern "C" void kernel_launch(void* const* d_in, const int* in_sizes, int n_in,
                              void* d_out, int out_size, void* d_ws, size_t ws_size,
                              hipStream_t stream) {
  const float* x   = (const float*)d_in[0];   // [N,128] f32
  const int*   bi  = (const int*)  d_in[1];   // [N] i32, sorted
  const float* w   = (const float*)d_in[2];   // [128]
  const float* bsp = (const float*)d_in[3];   // [128]
  const float* ms  = (const float*)d_in[4];   // [128]
  float* out = (float*)d_out;
  const int N = in_sizes[1];                  // node count

  graphnorm_kernel<<<GN_B, 512, 0, stream>>>(x, bi, w, bsp, ms, out, N);
}